// SMoE_15040975470629
// MI455X (gfx1250) — compile-verified
//
#include <hip/hip_runtime.h>
#include <hip/hip_bf16.h>
#include <math.h>

// ---------------------------------------------------------------------------
// SMoE forward for MI455X (gfx1250, wave32, WMMA, async-LDS staging).
// Shapes (fixed by reference): B=4 N=1024 Edim=1024 H=8 D=128 nE=16 Se=256 A=2
// ---------------------------------------------------------------------------

typedef __attribute__((ext_vector_type(2))) float v2f;
typedef __attribute__((ext_vector_type(8))) float v8f;
typedef __attribute__((ext_vector_type(4))) int   v4i_t;

#define T_TOK 4096
#define EDIM  1024
#define NHEAD 8
#define DHEAD 128
#define NEXP  16
#define SEXP  256
#define KS    16   // K-slab staged per step in the big GEMMs

// ---- gfx1250 async Global->LDS path (types learned from clang diagnostics) -
#if __has_builtin(__builtin_amdgcn_global_load_async_to_lds_b128) && \
    __has_builtin(__builtin_amdgcn_global_load_async_to_lds_b32)  && \
    __has_builtin(__builtin_amdgcn_s_wait_asynccnt)
#define HAS_ASYNC_LDS 1
#else
#define HAS_ASYNC_LDS 0
#endif

#if HAS_ASYNC_LDS
typedef __attribute__((address_space(1))) v4i_t* gp_v4i;
typedef __attribute__((address_space(3))) v4i_t* lp_v4i;
typedef __attribute__((address_space(1))) int*   gp_i32;
typedef __attribute__((address_space(3))) int*   lp_i32;

static __device__ __forceinline__ void async_ld_b128(const void* g, void* l) {
  __builtin_amdgcn_global_load_async_to_lds_b128(
      (gp_v4i)(uintptr_t)g, (lp_v4i)(unsigned int)(uintptr_t)l, 0, 0);
}
static __device__ __forceinline__ void async_ld_b32(const void* g, void* l) {
  __builtin_amdgcn_global_load_async_to_lds_b32(
      (gp_i32)(uintptr_t)g, (lp_i32)(unsigned int)(uintptr_t)l, 0, 0);
}
#endif

static __device__ __forceinline__ v8f v8f_zero() {
  v8f z = {0.f, 0.f, 0.f, 0.f, 0.f, 0.f, 0.f, 0.f};
  return z;
}

// D = A(16x4 f32) * B(4x16 f32) + C(16x16 f32), full-precision fp32 WMMA.
static __device__ __forceinline__ v8f wmma_f32(v2f a, v2f b, v8f c) {
  return __builtin_amdgcn_wmma_f32_16x16x4_f32(false, a, false, b,
                                               (short)0, c, false, false);
}

// ---------------------------------------------------------------------------
// Kernel 1: router logits + top-2 + sigmoid gates (dense g[t][e], 0 if unused)
// One wave32 per token.
// ---------------------------------------------------------------------------
__global__ __launch_bounds__(256) void router_top2_kernel(
    const float* __restrict__ x, const float* __restrict__ wr,
    float* __restrict__ gates) {
  const int lane = threadIdx.x & 31;
  const int tok  = (int)((blockIdx.x * blockDim.x + threadIdx.x) >> 5);
  const float* xp = x + (size_t)tok * EDIM;

  float lg[NEXP];
#pragma unroll
  for (int e = 0; e < NEXP; ++e) lg[e] = 0.f;

  for (int i = lane; i < EDIM; i += 32) {
    const float xv = xp[i];
#pragma unroll
    for (int e = 0; e < NEXP; ++e) lg[e] = fmaf(xv, wr[e * EDIM + i], lg[e]);
  }
#pragma unroll
  for (int off = 16; off >= 1; off >>= 1) {
#pragma unroll
    for (int e = 0; e < NEXP; ++e) lg[e] += __shfl_xor(lg[e], off, 32);
  }
  int b1 = 0;
#pragma unroll
  for (int e = 1; e < NEXP; ++e) if (lg[e] > lg[b1]) b1 = e;
  int b2 = (b1 == 0) ? 1 : 0;
#pragma unroll
  for (int e = 0; e < NEXP; ++e)
    if (e != b1 && lg[e] > lg[b2]) b2 = e;

  if (lane < NEXP) {
    float g = 0.f;
    if (lane == b1 || lane == b2) g = 1.f / (1.f + expf(-lg[lane]));
    gates[(size_t)tok * NEXP + lane] = g;
  }
}

// ---------------------------------------------------------------------------
// Kernel 2/4: C(T x 1024) = A(T x 1024) @ W(1024 x 1024)^T, fp32 WMMA.
// Block = 8 waves = 128 M x 64 N tile; W K-slabs (64 rows x 16 K) staged to
// LDS (async double-buffered when available), shared by all 8 waves.
// ---------------------------------------------------------------------------
#define LBS 20  // LDS row stride (floats): 80B rows -> 16B aligned, few conflicts
__global__ __launch_bounds__(256) void gemm_nt_wmma_kernel(
    const float* __restrict__ A, const float* __restrict__ W,
    float* __restrict__ C) {
  const int tid  = threadIdx.x;
  const int lane = tid & 31;
  const int lm   = lane & 15;
  const int lh   = lane >> 4;
  const int wv   = tid >> 5;
  const int mblk = (int)blockIdx.x >> 4;  // 0..T/128-1
  const int n64  = (int)blockIdx.x & 15;  // 0..15
  const int mbase = mblk * 128 + wv * 16;
  const int nbase = n64 * 64;

  __shared__ float lb[2][64 * LBS];

  // staging coords: 256 threads cover 64 rows x 16 K floats (4 floats each)
  const int sr  = tid >> 2;        // 0..63 (row within N-tile)
  const int sc4 = (tid & 3) * 4;   // 0,4,8,12 (col within K-slab)
  const float* wsrc = W + (size_t)(nbase + sr) * EDIM + sc4;

  v8f acc[4];
#pragma unroll
  for (int j = 0; j < 4; ++j) acc[j] = v8f_zero();

  const float* arow = A + (size_t)(mbase + lm) * EDIM + lh * 2;

#if HAS_ASYNC_LDS
  async_ld_b128(wsrc, &lb[0][sr * LBS + sc4]);
#endif
  for (int i = 0; i < EDIM / KS; ++i) {
    const int k0 = i * KS;
#if HAS_ASYNC_LDS
    if (i + 1 < EDIM / KS) {
      async_ld_b128(wsrc + k0 + KS, &lb[(i + 1) & 1][sr * LBS + sc4]);
      __builtin_amdgcn_s_wait_asynccnt(1);  // my slab i landed
    } else {
      __builtin_amdgcn_s_wait_asynccnt(0);
    }
    __syncthreads();                        // everyone's slab i landed
    const float* bb = lb[i & 1];
#else
    const float4 wreg = *(const float4*)(wsrc + k0);
    __syncthreads();                        // previous slab fully consumed
    *(float4*)&lb[0][sr * LBS + sc4] = wreg;
    __syncthreads();                        // slab i visible to all waves
    const float* bb = lb[0];
#endif
    __builtin_prefetch(arow + k0 + 4 * KS, 0, 1);  // global_prefetch_b8
#pragma unroll
    for (int kk = 0; kk < 4; ++kk) {
      const v2f a = *(const v2f*)(arow + k0 + kk * 4);
#pragma unroll
      for (int j = 0; j < 4; ++j) {
        const v2f b = *(const v2f*)&bb[(j * 16 + lm) * LBS + kk * 4 + lh * 2];
        acc[j] = wmma_f32(a, b, acc[j]);
      }
    }
#if HAS_ASYNC_LDS
    __syncthreads();  // all waves done reading lb[i&1] before it is re-staged
#endif
  }
  // C layout: VGPR i holds row m = i + 8*(lane>=16), col n = lane&15
#pragma unroll
  for (int j = 0; j < 4; ++j) {
#pragma unroll
    for (int i = 0; i < 8; ++i) {
      C[(size_t)(mbase + i + 8 * lh) * EDIM + nbase + j * 16 + lm] = acc[j][i];
    }
  }
}

// ---------------------------------------------------------------------------
// Kernel 3: per-head expert FFN, gate-masked dense sweep over experts.
// Block = 8 waves = (head h, 128 tokens); all waves uniform so the block can
// cooperatively stage Ve^T slabs (16 Se x 128 D, transposed) into LDS.
//   y[tile] = sum_e diag(g_e) * gelu(Q @ Ke^T) @ Ve
// ---------------------------------------------------------------------------
#define VTS 18  // vT row stride (floats): 72B rows -> 8B aligned, conflict-free
__global__ __launch_bounds__(256) void expert_moe_kernel(
    const float* __restrict__ q,      // (T, H*D) : q[t][h*128+d]
    const float* __restrict__ kf,     // (H, nE, Se, D)
    const float* __restrict__ vf,     // (H, nE, Se, D)
    const float* __restrict__ gates,  // (T, nE)
    float* __restrict__ y) {          // (T, H*D)
  const int tid  = threadIdx.x;
  const int lane = tid & 31;
  const int lm   = lane & 15;
  const int lh   = lane >> 4;
  const int wv   = tid >> 5;
  const int h    = (int)blockIdx.x >> 5;  // 0..7
  const int mblk = (int)blockIdx.x & 31;  // 0..31
  const int tbase = mblk * 128 + wv * 16;

  __shared__ float vT[DHEAD * VTS];      // transposed V slab: vT[d][s]
  __shared__ float lds_act[8][16 * 17];  // per-wave 16x16 act tile
  float* act = lds_act[wv];

  // staging coords: 256 threads cover 16 s-rows x 128 d (8 floats each)
  const int ss = tid >> 4;         // 0..15 (s within slab)
  const int d0 = (tid & 15) * 8;   // 0..120

  v8f acc[8];
#pragma unroll
  for (int nt = 0; nt < 8; ++nt) acc[nt] = v8f_zero();

  const float* qbase = q + (size_t)(tbase + lm) * EDIM + h * DHEAD + lh * 2;

  for (int e = 0; e < NEXP; ++e) {
    // per-row gates for the rows this lane owns in the C layout (>=0, 0=off)
    float g8[8];
#pragma unroll
    for (int i = 0; i < 8; ++i)
      g8[i] = gates[(size_t)(tbase + i + 8 * lh) * NEXP + e];

    const float* Ke = kf + (size_t)(h * NEXP + e) * SEXP * DHEAD;
    const float* Ve = vf + (size_t)(h * NEXP + e) * SEXP * DHEAD;

    for (int sc = 0; sc < SEXP / 16; ++sc) {
      // ---- cooperative transpose-stage of Ve slab into vT[d][s] ----
      const float* vsrc = Ve + (size_t)(sc * 16 + ss) * DHEAD + d0;
      __syncthreads();  // everyone done reading previous vT slab
#if HAS_ASYNC_LDS
#pragma unroll
      for (int u = 0; u < 8; ++u)
        async_ld_b32(vsrc + u, &vT[(d0 + u) * VTS + ss]);
      __builtin_amdgcn_s_wait_asynccnt(0);
#else
      {
        const float4 v0 = *(const float4*)(vsrc);
        const float4 v1 = *(const float4*)(vsrc + 4);
        vT[(d0 + 0) * VTS + ss] = v0.x;
        vT[(d0 + 1) * VTS + ss] = v0.y;
        vT[(d0 + 2) * VTS + ss] = v0.z;
        vT[(d0 + 3) * VTS + ss] = v0.w;
        vT[(d0 + 4) * VTS + ss] = v1.x;
        vT[(d0 + 5) * VTS + ss] = v1.y;
        vT[(d0 + 6) * VTS + ss] = v1.z;
        vT[(d0 + 7) * VTS + ss] = v1.w;
      }
#endif
      __syncthreads();  // vT slab visible to all waves

      // ---- scores tile: S(16x16) = Q(16x128) @ Ke[sc-block]^T ----
      v8f s = v8f_zero();
      const float* krow = Ke + (size_t)(sc * 16 + lm) * DHEAD + lh * 2;
#pragma unroll
      for (int k0 = 0; k0 < DHEAD; k0 += 4) {
        const v2f a = *(const v2f*)(qbase + k0);
        const v2f b = *(const v2f*)(krow + k0);
        s = wmma_f32(a, b, s);
      }
      // ---- exact gelu + gate, write act tile to per-wave LDS ----
#pragma unroll
      for (int i = 0; i < 8; ++i) {
        const float v  = s[i];
        const float gl = 0.5f * v * (1.f + erff(v * 0.70710678118654752f));
        act[(i + 8 * lh) * 17 + lm] = gl * g8[i];
      }
      __builtin_amdgcn_wave_barrier();  // keep LDS store->load ordering
      // ---- y(16x128) += act(16x16) @ Ve[sc-block](16x128) via vT ----
#pragma unroll
      for (int kk = 0; kk < 4; ++kk) {
        v2f a2;
        a2.x = act[lm * 17 + kk * 4 + lh * 2];
        a2.y = act[lm * 17 + kk * 4 + lh * 2 + 1];
        const int c = kk * 4 + lh * 2;  // s-local column
#pragma unroll
        for (int nt = 0; nt < 8; ++nt) {
          const v2f b2 = *(const v2f*)&vT[(nt * 16 + lm) * VTS + c];
          acc[nt] = wmma_f32(a2, b2, acc[nt]);
        }
      }
      __builtin_amdgcn_wave_barrier();
    }
  }
  // store this wave's private 16x128 slab of y
#pragma unroll
  for (int nt = 0; nt < 8; ++nt) {
#pragma unroll
    for (int i = 0; i < 8; ++i) {
      y[(size_t)(tbase + i + 8 * lh) * EDIM + h * DHEAD + nt * 16 + lm] =
          acc[nt][i];
    }
  }
}

// ---------------------------------------------------------------------------
extern "C" void kernel_launch(void* const* d_in, const int* in_sizes, int n_in,
                              void* d_out, int out_size, void* d_ws,
                              size_t ws_size, hipStream_t stream) {
  (void)in_sizes; (void)n_in; (void)out_size; (void)ws_size;
  const float* x        = (const float*)d_in[0];  // (4,1024,1024)
  const float* w_router = (const float*)d_in[1];  // (16,1024)
  const float* w_mh     = (const float*)d_in[2];  // (1024,1024)
  const float* w_mg     = (const float*)d_in[3];  // (1024,1024)
  const float* kf       = (const float*)d_in[4];  // (8,16,256,128)
  const float* vf       = (const float*)d_in[5];  // (8,16,256,128)
  float* out = (float*)d_out;                     // (4,1024,1024)

  // workspace: q (16MB) | y (16MB) | gates (256KB)
  float* q     = (float*)d_ws;
  float* yb    = q + (size_t)T_TOK * EDIM;
  float* gates = yb + (size_t)T_TOK * EDIM;

  // 1) router + top-2 + sigmoid gates : 4096 waves
  router_top2_kernel<<<(T_TOK * 32) / 256, 256, 0, stream>>>(x, w_router, gates);

  // 2) q = x @ w_mh^T : 512 blocks of 8 waves (128M x 64N tiles)
  gemm_nt_wmma_kernel<<<(T_TOK / 128) * (EDIM / 64), 256, 0, stream>>>(
      x, w_mh, q);

  // 3) expert FFN : H * (T/128) = 256 blocks of 8 waves
  expert_moe_kernel<<<NHEAD * (T_TOK / 128), 256, 0, stream>>>(
      q, kf, vf, gates, yb);

  // 4) out = y @ w_mg^T
  gemm_nt_wmma_kernel<<<(T_TOK / 128) * (EDIM / 64), 256, 0, stream>>>(
      yb, w_mg, out);
}